// GredRanking_Loss_66168266162903
// MI455X (gfx1250) — compile-verified
//
#include <hip/hip_runtime.h>

// ---------------------------------------------------------------------------
// GredRanking loss for MI455X (gfx1250).
// Bandwidth-bound (<1 FLOP/byte) => no WMMA; the CDNA5-specific path used is
// the async global->LDS data mover (ASYNCcnt) for the stencil pass, plus
// wave32 shfl reductions and f64 global atomics.
// ---------------------------------------------------------------------------

#define DEVINL __device__ __forceinline__

constexpr int   B_    = 16;
constexpr int   H_    = 512;
constexpr int   W_    = 1024;
constexpr int   N_    = H_ * W_;           // 524288 per batch
constexpr int   SCAN_CHUNK = 4096;         // elements per scan block (256 thr x 16)
constexpr int   NBLK  = N_ / SCAN_CHUNK;   // 128 scan blocks per batch
constexpr float THETA_ = 0.02f;

// ---- gfx1250 async global -> LDS copy (16 bytes per lane) ------------------
typedef int v4i_ __attribute__((vector_size(16)));
typedef __attribute__((address_space(1))) v4i_* gv4i_ptr;   // global
typedef __attribute__((address_space(3))) v4i_* lv4i_ptr;   // LDS

DEVINL void async_copy_b128(const float* gsrc, float* lds_dst) {
#if defined(__gfx1250__) && __has_builtin(__builtin_amdgcn_global_load_async_to_lds_b128)
  __builtin_amdgcn_global_load_async_to_lds_b128(
      (gv4i_ptr)(void*)gsrc,
      (lv4i_ptr)(void*)lds_dst,
      0, 0);
#elif defined(__gfx1250__)
  unsigned laddr =
      (unsigned)(unsigned long long)(__attribute__((address_space(3))) char*)(void*)lds_dst;
  unsigned long long gaddr = (unsigned long long)gsrc;
  asm volatile("global_load_async_to_lds_b128 %0, %1, off"
               :: "v"(laddr), "v"(gaddr) : "memory");
#else
  // host-pass / non-gfx1250 parse fallback
  ((float4*)lds_dst)[0] = ((const float4*)gsrc)[0];
#endif
}

DEVINL void wait_async0() {
#if defined(__gfx1250__) && __has_builtin(__builtin_amdgcn_s_wait_asynccnt)
  __builtin_amdgcn_s_wait_asynccnt(0);
#elif defined(__gfx1250__)
  asm volatile("s_wait_asynccnt 0" ::: "memory");
#endif
}

// ---- wave32 reductions -----------------------------------------------------
DEVINL float  wredF(float v)  { for (int d = 16; d > 0; d >>= 1) v += __shfl_xor(v, d, 32); return v; }
DEVINL int    wredI(int v)    { for (int d = 16; d > 0; d >>= 1) v += __shfl_xor(v, d, 32); return v; }
DEVINL double wredD(double v) { for (int d = 16; d > 0; d >>= 1) v += __shfl_xor(v, d, 32); return v; }
DEVINL unsigned wredU(unsigned v){ for (int d = 16; d > 0; d >>= 1) v += __shfl_xor(v, d, 32); return v; }

// ===========================================================================
// K0: zero the small accumulators
// ===========================================================================
__global__ void k0_init(double* twsumG, double* sums, unsigned* cnts) {
  int t = threadIdx.x;
  if (t == 0) twsumG[0] = 0.0;
  if (t < 2) { sums[t] = 0.0; cnts[t] = 0u; }
}

// ===========================================================================
// K1: texture weight tw + global sum(tw).
// One block per (row y, batch b). LDS holds rows y and y+1 of all 3 channels,
// staged with async global->LDS B128 copies (256 lanes x 16B = full 1024-wide
// row per (row,channel)).
// ===========================================================================
__global__ void k1_tw(const float* __restrict__ img, float* __restrict__ tw,
                      double* __restrict__ twsumG) {
  const int y = blockIdx.x;
  const int b = blockIdx.y;
  const int t = threadIdx.x;              // 256 threads = 8 waves
  const int y1 = (y + 1 < H_) ? y + 1 : y;
  const bool has_gy = (y + 1 < H_);

  __shared__ float sm[2][3][W_];          // 24 KB
  __shared__ float wpart[8];

#pragma unroll
  for (int c = 0; c < 3; ++c) {
    const float* g0 = img + (((size_t)b * 3 + c) * H_ + y ) * W_ + t * 4;
    const float* g1 = img + (((size_t)b * 3 + c) * H_ + y1) * W_ + t * 4;
    async_copy_b128(g0, &sm[0][c][t * 4]);
    async_copy_b128(g1, &sm[1][c][t * 4]);
  }
  wait_async0();
  __syncthreads();

  const size_t rowbase = ((size_t)b * H_ + y) * W_;   // index into tw (B,N), n = y*W + x
  float acc = 0.f;
#pragma unroll
  for (int j = 0; j < 4; ++j) {
    const int x  = t + 256 * j;
    const int xp = (x + 1 < W_) ? x + 1 : x;
    float gx = (fabsf(sm[0][0][x] - sm[0][0][xp]) +
                fabsf(sm[0][1][x] - sm[0][1][xp]) +
                fabsf(sm[0][2][x] - sm[0][2][xp])) * (1.f / 3.f);
    float gy = (fabsf(sm[0][0][x] - sm[1][0][x]) +
                fabsf(sm[0][1][x] - sm[1][1][x]) +
                fabsf(sm[0][2][x] - sm[1][2][x])) * (1.f / 3.f);
    float v = 0.f;
    if (x + 1 < W_) v += gx;
    if (has_gy)     v += gy;
    v *= 0.5f;
    tw[rowbase + x] = v;
    acc += v;
  }

  // block reduction -> one f64 atomic per block
  acc = wredF(acc);
  const int lane = t & 31, wid = t >> 5;
  if (lane == 0) wpart[wid] = acc;
  __syncthreads();
  if (wid == 0) {
    float v = (lane < 8) ? wpart[lane] : 0.f;
    v = wredF(v);
    if (lane == 0) atomicAdd(twsumG, (double)v);
  }
}

// ===========================================================================
// K2: per-scan-block count of valid (tw >= global mean) elements
// ===========================================================================
__global__ void k2_count(const float* __restrict__ tw, const double* __restrict__ twsumG,
                         int* __restrict__ blkcnt) {
  const int blk = blockIdx.x, b = blockIdx.y, t = threadIdx.x;
  const float mean = (float)(twsumG[0] / (double)((size_t)B_ * N_));
  const size_t base = (size_t)b * N_ + (size_t)blk * SCAN_CHUNK + (size_t)t * 16;

  int cnt = 0;
#pragma unroll
  for (int i = 0; i < 16; ++i) cnt += (tw[base + i] >= mean) ? 1 : 0;

  cnt = wredI(cnt);
  __shared__ int wpart[8];
  const int lane = t & 31, wid = t >> 5;
  if (lane == 0) wpart[wid] = cnt;
  __syncthreads();
  if (wid == 0) {
    int v = (lane < 8) ? wpart[lane] : 0;
    v = wredI(v);
    if (lane == 0) blkcnt[b * NBLK + blk] = v;
  }
}

// ===========================================================================
// K3: per-batch exclusive scan of the NBLK block counts + nval
// one block per batch, NBLK threads
// ===========================================================================
__global__ void k3_scan(const int* __restrict__ blkcnt, int* __restrict__ blkoff,
                        int* __restrict__ nval) {
  const int b = blockIdx.x, t = threadIdx.x;
  __shared__ int s[NBLK];
  const int v = blkcnt[b * NBLK + t];
  s[t] = v;
  __syncthreads();
  for (int d = 1; d < NBLK; d <<= 1) {
    int x = (t >= d) ? s[t - d] : 0;
    __syncthreads();
    s[t] += x;
    __syncthreads();
  }
  blkoff[b * NBLK + t] = s[t] - v;          // exclusive
  if (t == NBLK - 1) nval[b] = s[t];
}

// ===========================================================================
// K4: scatter positions of valid elements into compacted pos[] array
// (turns searchsorted into a single gather in K5)
// ===========================================================================
__global__ void k4_scatter(const float* __restrict__ tw, const double* __restrict__ twsumG,
                           const int* __restrict__ blkoff, int* __restrict__ pos) {
  const int blk = blockIdx.x, b = blockIdx.y, t = threadIdx.x;
  const float mean = (float)(twsumG[0] / (double)((size_t)B_ * N_));
  const int    nloc = blk * SCAN_CHUNK + t * 16;        // n within batch
  const size_t base = (size_t)b * N_ + nloc;

  float v[16];
  int cnt = 0;
#pragma unroll
  for (int i = 0; i < 16; ++i) { v[i] = tw[base + i]; cnt += (v[i] >= mean) ? 1 : 0; }

  // inclusive wave scan of per-thread counts
  const int lane = t & 31, wid = t >> 5;
  int sc = cnt;
#pragma unroll
  for (int d = 1; d < 32; d <<= 1) { int x = __shfl_up(sc, d, 32); if (lane >= d) sc += x; }

  __shared__ int wsum[8];
  if (lane == 31) wsum[wid] = sc;
  __syncthreads();
  int wbase = 0;
  for (int i = 0; i < wid; ++i) wbase += wsum[i];

  int rank = blkoff[b * NBLK + blk] + wbase + (sc - cnt);
  const size_t bb = (size_t)b * N_;
#pragma unroll
  for (int i = 0; i < 16; ++i) {
    if (v[i] >= mean) { pos[bb + rank] = nloc + i; ++rank; }
  }
}

// ===========================================================================
// K5: fused sampling + ranking loss. Gathers of gt/pred/pos land inside a
// 2 MB per-batch window -> L2-resident on MI455X (192 MB L2).
// ===========================================================================
__global__ void k5_loss(const float* __restrict__ pred, const float* __restrict__ gt,
                        const float* __restrict__ tw,   const float* __restrict__ u_rand,
                        const int* __restrict__ pos,    const int* __restrict__ nval,
                        const double* __restrict__ twsumG,
                        double* __restrict__ sums, unsigned* __restrict__ cnts) {
  const float mean = (float)(twsumG[0] / (double)((size_t)B_ * N_));
  const unsigned long long total  = (unsigned long long)B_ * N_;
  const unsigned long long stride = (unsigned long long)gridDim.x * blockDim.x;

  double lsum = 0.0, ssum = 0.0;
  unsigned ncnt = 0u, zcnt = 0u;

  for (unsigned long long g = (unsigned long long)blockIdx.x * blockDim.x + threadIdx.x;
       g < total; g += stride) {
    const int b = (int)(g / (unsigned)N_);
    const bool invalid = tw[g] < mean;
    if (!invalid) continue;                 // skip gathers for valid pixels

    const int nv = nval[b];
    int u = (int)(u_rand[g] * (float)nv);   // truncation, as in astype(int32)
    if (u > nv - 1) u = nv - 1;
    if (u < 0)      u = 0;
    const size_t bb  = (size_t)b * N_;
    const int    idx = pos[bb + u];

    const float gtn = gt[g],  gtv = gt[bb + idx];
    const float prn = pred[g], prv = pred[bb + idx];
    const float r1 = gtn / gtv, r2 = gtv / gtn;
    const float target = (r1 >= 1.f + THETA_) ? 1.f : ((r2 > 1.f + THETA_) ? -1.f : 0.f);
    const float d = prn - prv;

    if (target != 0.f) {
      lsum += (double)log1pf(expf(-target * d));
      ++ncnt;
    } else {
      ssum += (double)(d * d);
      ++zcnt;
    }
  }

  // block reduce -> global f64 / u32 atomics
  lsum = wredD(lsum); ssum = wredD(ssum);
  ncnt = wredU(ncnt); zcnt = wredU(zcnt);

  __shared__ double sdl[8], sds[8];
  __shared__ unsigned snl[8], snz[8];
  const int lane = threadIdx.x & 31, wid = threadIdx.x >> 5;
  if (lane == 0) { sdl[wid] = lsum; sds[wid] = ssum; snl[wid] = ncnt; snz[wid] = zcnt; }
  __syncthreads();
  if (wid == 0) {
    double a = (lane < 8) ? sdl[lane] : 0.0;
    double c = (lane < 8) ? sds[lane] : 0.0;
    unsigned e = (lane < 8) ? snl[lane] : 0u;
    unsigned f = (lane < 8) ? snz[lane] : 0u;
    a = wredD(a); c = wredD(c); e = wredU(e); f = wredU(f);
    if (lane == 0) {
      atomicAdd(&sums[0], a);
      atomicAdd(&sums[1], c);
      atomicAdd(&cnts[0], e);
      atomicAdd(&cnts[1], f);
    }
  }
}

// ===========================================================================
// K6: finalize scalar output
// ===========================================================================
__global__ void k6_fin(const double* __restrict__ sums, const unsigned* __restrict__ cnts,
                       float* __restrict__ out) {
  const float ll = (float)sums[0] / (float)cnts[0];
  const float sl = (float)sums[1] / (float)cnts[1];
  out[0] = ll + sl;
}

// ===========================================================================
extern "C" void kernel_launch(void* const* d_in, const int* in_sizes, int n_in,
                              void* d_out, int out_size, void* d_ws, size_t ws_size,
                              hipStream_t stream) {
  (void)in_sizes; (void)n_in; (void)out_size; (void)ws_size;

  const float* pred   = (const float*)d_in[0];   // (B,1,H,W)
  const float* gt     = (const float*)d_in[1];   // (B,1,H,W)
  const float* img    = (const float*)d_in[2];   // (B,3,H,W)
  const float* u_rand = (const float*)d_in[3];   // (B,N)
  float* out = (float*)d_out;

  // ---- workspace layout (bytes) ----
  constexpr size_t BN = (size_t)B_ * N_;
  char* ws = (char*)d_ws;
  float*  tw      = (float*)(ws);                      // BN*4
  int*    pos     = (int*)  (ws + BN * 4);             // BN*4
  char*   ctl     = ws + 2 * BN * 4;
  double* twsumG  = (double*)(ctl + 0);                // 8
  double* sums    = (double*)(ctl + 8);                // 16
  unsigned* cnts  = (unsigned*)(ctl + 24);             // 8
  int*    nval    = (int*)(ctl + 32);                  // 64
  int*    blkcnt  = (int*)(ctl + 96);                  // B*NBLK*4
  int*    blkoff  = (int*)(ctl + 96 + (size_t)B_ * NBLK * 4);

  k0_init   <<<1, 32, 0, stream>>>(twsumG, sums, cnts);
  k1_tw     <<<dim3(H_, B_), 256, 0, stream>>>(img, tw, twsumG);
  k2_count  <<<dim3(NBLK, B_), 256, 0, stream>>>(tw, twsumG, blkcnt);
  k3_scan   <<<B_, NBLK, 0, stream>>>(blkcnt, blkoff, nval);
  k4_scatter<<<dim3(NBLK, B_), 256, 0, stream>>>(tw, twsumG, blkoff, pos);
  k5_loss   <<<4096, 256, 0, stream>>>(pred, gt, tw, u_rand, pos, nval, twsumG, sums, cnts);
  k6_fin    <<<1, 1, 0, stream>>>(sums, cnts, out);
}